// QAttn_24129126269083
// MI455X (gfx1250) — compile-verified
//
#include <hip/hip_runtime.h>
#include <hip/hip_bf16.h>
#include <cstddef>
#include <cstdint>

// ---------------------------------------------------------------------------
// Quantized grouped conv (16 groups, 4-in/4-out per group) via CDNA5 WMMA.
//   D(16 och x 16 pix) += A(16x4, block-diag group weights) * B(4x16, pixels)
// using V_WMMA_F32_16X16X4_F32, accumulating over taps x 4 sub-groups.
// Input strips staged to LDS with GLOBAL_LOAD_ASYNC_TO_LDS_B128 (ASYNCcnt).
// ---------------------------------------------------------------------------

typedef __attribute__((ext_vector_type(2))) float v2f;
typedef __attribute__((ext_vector_type(8))) float v8f;
typedef __attribute__((ext_vector_type(4))) float v4f;
typedef __attribute__((ext_vector_type(4))) int   v4i;

typedef __attribute__((address_space(1))) v4i as1_v4i;
typedef __attribute__((address_space(3))) v4i as3_v4i;

#define HH 64
#define WW 64
#define CTOT 256
#define NCH 16   // channels per "quad" (4 groups x 4 channels)

#if defined(__has_builtin)
#if __has_builtin(__builtin_amdgcn_global_load_async_to_lds_b128)
#define HAVE_ASYNC_LDS 1
#endif
#endif

__device__ __forceinline__ void wait_asynccnt0() {
#if __has_builtin(__builtin_amdgcn_s_wait_asynccnt)
    __builtin_amdgcn_s_wait_asynccnt(0);
#else
    asm volatile("s_wait_asynccnt 0x0" ::: "memory");
#endif
}

template <int KH, int KW, int PH, int PW, int TW>
__global__ __launch_bounds__(256) void qgconv_wmma(
    const float* __restrict__ x,   // (64, 256, 64, 64), section at in_ch_off
    const float* __restrict__ w,   // (64, 4, KH, KW)
    const float* __restrict__ bias,// (64,)
    const float* __restrict__ s,   // scalar
    float* __restrict__ out,       // (64, 256, 64, 64), section at out_ch_off
    int in_ch_off, int out_ch_off)
{
    constexpr int TAPS = KH * KW;
    constexpr int LP   = (PW + 3) & ~3;       // left pad, 16B aligned interior
    constexpr int RS   = (LP + TW + PW + 3) & ~3;  // LDS row stride (floats)
    constexpr int WT   = WW / TW;             // w-tiles per image row
    constexpr int XS_ELEMS = NCH * HH * RS;

    extern __shared__ float lds[];
    float* xs = lds;                       // [NCH][HH][RS]
    float* wq = lds + XS_ELEMS;            // [16 och][4 ci][TAPS] quantized
    float* bq = wq + NCH * 4 * TAPS;       // [16]

    const int blk  = blockIdx.x;
    const int wt   = blk % WT;
    const int quad = (blk / WT) % 4;
    const int n    = blk / (WT * 4);
    const int w0   = wt * TW;

    const int tid  = threadIdx.x;
    const int lane = tid & 31;
    const int wave = tid >> 5;
    const int l16  = lane >> 4;   // 0 | 1  (half-wave)
    const int ln   = lane & 15;   // col within half-wave

    // ---- stage + fake-quantize weights (RNE round like jnp.round) ----
    const float scale = s[0];
    for (int i = tid; i < NCH * 4 * TAPS; i += 256) {
        const int oc = i / (4 * TAPS);
        const int rem = i % (4 * TAPS);
        const float wv = w[(quad * NCH + oc) * (4 * TAPS) + rem];
        float r = rintf(wv / scale);
        r = fminf(fmaxf(r, -8.0f), 7.0f);
        wq[i] = r * scale;
    }
    if (tid < NCH) bq[tid] = bias[quad * NCH + tid];

    // ---- stage input strip: 16 ch x 64 rows x TW interior + 2*PW halo ----
    const size_t xbase = ((size_t)n * CTOT + in_ch_off + quad * NCH) * (HH * WW);

    // interior: async b128 global -> LDS (both sides 16B aligned)
    for (int i = tid; i < NCH * HH * (TW / 4); i += 256) {
        const int ci = i / (HH * (TW / 4));
        const int rr = i % (HH * (TW / 4));
        const int hh = rr / (TW / 4);
        const int c4 = rr % (TW / 4);
        const float* src = x + xbase + (size_t)ci * (HH * WW) + hh * WW + w0 + 4 * c4;
        float* dst = xs + (ci * HH + hh) * RS + LP + 4 * c4;
#ifdef HAVE_ASYNC_LDS
        __builtin_amdgcn_global_load_async_to_lds_b128(
            (as1_v4i*)src, (as3_v4i*)dst, 0, 0);
#else
        *(v4f*)dst = *(const v4f*)src;
#endif
    }
    // halo columns (PW each side), zero outside image
    if constexpr (PW > 0) {
        for (int i = tid; i < NCH * HH * 2 * PW; i += 256) {
            const int ci = i / (HH * 2 * PW);
            const int rr = i % (HH * 2 * PW);
            const int hh = rr / (2 * PW);
            const int e  = rr % (2 * PW);
            const bool left = e < PW;
            const int col = left ? (LP - PW + e) : (LP + TW + (e - PW));
            const int gw  = left ? (w0 - PW + e) : (w0 + TW + (e - PW));
            float v = 0.0f;
            if (gw >= 0 && gw < WW)
                v = x[xbase + (size_t)ci * (HH * WW) + hh * WW + gw];
            xs[(ci * HH + hh) * RS + col] = v;
        }
    }
#ifdef HAVE_ASYNC_LDS
    wait_asynccnt0();
#endif
    __syncthreads();

    // ---- hoist A fragments (tile-invariant) into registers ----
    // A(16x4): row m=ln, K = v + 2*l16; nonzero only for rows of sub-group sb.
    v2f afrag[TAPS * 4];
    {
        const float* wrow = wq + ln * 4 * TAPS + 2 * l16 * TAPS;
        #pragma unroll
        for (int tap = 0; tap < TAPS; ++tap) {
            const float wk0 = wrow[tap];
            const float wk1 = wrow[TAPS + tap];
            #pragma unroll
            for (int sb = 0; sb < 4; ++sb) {
                const float sel = ((ln >> 2) == sb) ? 1.0f : 0.0f;
                afrag[tap * 4 + sb].x = sel * wk0;
                afrag[tap * 4 + sb].y = sel * wk1;
            }
        }
    }
    float breg[8];
    #pragma unroll
    for (int r = 0; r < 8; ++r) breg[r] = bq[r + 8 * l16];

    // ---- WMMA compute: each wave owns D tiles of 16 och x 16 pixels ----
    constexpr int PTILES = TW / 16;
    constexpr int NTILES = HH * PTILES;

    for (int t = wave; t < NTILES; t += 8) {
        const int hh = t / PTILES;
        const int pt = t % PTILES;
        v8f acc = {};
        #pragma unroll
        for (int kh = 0; kh < KH; ++kh) {
            const int hsrc = hh + kh - PH;
            if (hsrc >= 0 && hsrc < HH) {          // wave-uniform branch
                #pragma unroll
                for (int kw = 0; kw < KW; ++kw) {
                    const int tap = kh * KW + kw;
                    const int col = LP + pt * 16 + kw - PW + ln;
                    #pragma unroll
                    for (int sb = 0; sb < 4; ++sb) {
                        // B(4x16): K = v + 2*l16 (mirrors A), col = pixel
                        const float* bp =
                            &xs[((sb * 4 + 2 * l16) * HH + hsrc) * RS + col];
                        v2f bf;
                        bf.x = bp[0];
                        bf.y = bp[HH * RS];
                        acc = __builtin_amdgcn_wmma_f32_16x16x4_f32(
                            false, afrag[tap * 4 + sb], false, bf,
                            (short)0, acc, false, false);
                    }
                }
            }
        }
        // D: vgpr r -> och m = r + 8*l16, pixel = ln. Add bias, store.
        const size_t obase = ((size_t)n * CTOT + out_ch_off + quad * NCH) * (HH * WW)
                           + (size_t)hh * WW + w0 + pt * 16 + ln;
        #pragma unroll
        for (int r = 0; r < 8; ++r) {
            const int m = r + 8 * l16;
            out[obase + (size_t)m * (HH * WW)] = acc[r] + breg[r];
        }
    }
}

// Identity channels: out[:, :64] = x[:, :64]  (float4 copy)
__global__ __launch_bounds__(256) void copy_id(const float* __restrict__ x,
                                               float* __restrict__ out)
{
    const size_t i = (size_t)blockIdx.x * blockDim.x + threadIdx.x;   // float4 idx
    constexpr size_t PER_N  = (size_t)64 * HH * WW / 4;               // id chans / batch
    constexpr size_t STRIDE = (size_t)CTOT * HH * WW / 4;             // full batch
    const size_t n = i / PER_N;
    const size_t r = i % PER_N;
    const v4f* __restrict__ xv = (const v4f*)x;
    v4f* __restrict__ ov = (v4f*)out;
    ov[n * STRIDE + r] = xv[n * STRIDE + r];
}

extern "C" void kernel_launch(void* const* d_in, const int* in_sizes, int n_in,
                              void* d_out, int out_size, void* d_ws, size_t ws_size,
                              hipStream_t stream)
{
    (void)in_sizes; (void)n_in; (void)out_size; (void)d_ws; (void)ws_size;
    const float* x    = (const float*)d_in[0];
    const float* w_hw = (const float*)d_in[1];
    const float* b_hw = (const float*)d_in[2];
    const float* s_hw = (const float*)d_in[3];
    const float* w_w  = (const float*)d_in[4];
    const float* b_w  = (const float*)d_in[5];
    const float* s_w  = (const float*)d_in[6];
    const float* w_h  = (const float*)d_in[7];
    const float* b_h  = (const float*)d_in[8];
    const float* s_h  = (const float*)d_in[9];
    float* out = (float*)d_out;

    // LDS footprints (bytes); CDNA5 allows up to 320KB/workgroup.
    // y_h : RS=16 -> 16*64*16  floats;  y_w: RS=48 -> 16*64*48;  y_hw: RS=40
    const int sh_h  = (16 * 64 * 16 + 16 * 4 * 11 + 16) * 4;   //  68,416
    const int sh_w  = (16 * 64 * 48 + 16 * 4 * 11 + 16) * 4;   // 199,488
    const int sh_hw = (16 * 64 * 40 + 16 * 4 * 9  + 16) * 4;   // 166,208
    (void)hipFuncSetAttribute((const void*)qgconv_wmma<11, 1, 5, 0, 16>,
                              hipFuncAttributeMaxDynamicSharedMemorySize, sh_h);
    (void)hipFuncSetAttribute((const void*)qgconv_wmma<1, 11, 0, 5, 32>,
                              hipFuncAttributeMaxDynamicSharedMemorySize, sh_w);
    (void)hipFuncSetAttribute((const void*)qgconv_wmma<3, 3, 1, 1, 32>,
                              hipFuncAttributeMaxDynamicSharedMemorySize, sh_hw);

    // identity: 64 batches x 64ch x 4096 px / 4 = 4,194,304 float4s
    copy_id<<<16384, 256, 0, stream>>>(x, out);

    // y_h : 11x1, pad (5,0), in/out channel offset 64
    qgconv_wmma<11, 1, 5, 0, 16><<<64 * 4 * 4, 256, sh_h, stream>>>(
        x, w_h, b_h, s_h, out, 64, 64);
    // y_w : 1x11, pad (0,5), offset 128
    qgconv_wmma<1, 11, 0, 5, 32><<<64 * 4 * 2, 256, sh_w, stream>>>(
        x, w_w, b_w, s_w, out, 128, 128);
    // y_hw: 3x3, pad (1,1), offset 192
    qgconv_wmma<3, 3, 1, 1, 32><<<64 * 4 * 2, 256, sh_hw, stream>>>(
        x, w_hw, b_hw, s_hw, out, 192, 192);
}